// s4MambaModule_29420525977714
// MI455X (gfx1250) — compile-verified
//
#include <hip/hip_runtime.h>
#include <math.h>

// Problem constants (match reference)
#define L_SEQ   4096
#define DM      1024
#define DI      2048
#define NFFT    8192
#define NBATCH  2

typedef __attribute__((ext_vector_type(2))) float        v2f;
typedef __attribute__((ext_vector_type(8))) float        v8f;
typedef __attribute__((ext_vector_type(4))) unsigned int u32x4;
typedef __attribute__((ext_vector_type(8))) int          i32x8;
typedef __attribute__((ext_vector_type(4))) int          i32x4;

// ---------------------------------------------------------------------------
// GEMM 1: xz[b,e,l] = sum_d hs[b,l,d] * Win[e,d]
// M=(b,l)=8192, N=e=4096, K=d=1024.  Per-wave 16x64 tile via 4 WMMA accums.
// ---------------------------------------------------------------------------
__global__ __launch_bounds__(256) void gemm_xz_kernel(
    const float* __restrict__ hs, const float* __restrict__ Win,
    float* __restrict__ xbuf, float* __restrict__ zbuf)
{
  const int lane = threadIdx.x & 31;
  const int wave = threadIdx.x >> 5;
  const int gw   = blockIdx.x * 8 + wave;
  const int ng   = gw & 63;            // 4096/64 = 64 n-groups
  const int mt   = gw >> 6;            // 512 m tiles
  const int m0   = mt << 4;
  const int n0   = ng << 6;
  const int koff = (lane >> 4) << 1;   // lanes 16-31 hold K=2,3
  const int lr   = lane & 15;

  const float* pa = hs  + (m0 + lr) * DM + koff;
  const float* pb = Win + (n0 + lr) * DM + koff;
  __builtin_prefetch(pa, 0, 0);        // global_prefetch_b8
  __builtin_prefetch(pb, 0, 0);

  v8f acc[4] = {v8f{}, v8f{}, v8f{}, v8f{}};
  for (int k0 = 0; k0 < DM; k0 += 4) {
    v2f a = *(const v2f*)(pa + k0);
#pragma unroll
    for (int t = 0; t < 4; ++t) {
      v2f b = *(const v2f*)(pb + t * 16 * DM + k0);
      acc[t] = __builtin_amdgcn_wmma_f32_16x16x4_f32(
          false, a, false, b, (short)0, acc[t], false, false);
    }
  }
#pragma unroll
  for (int r = 0; r < 8; ++r) {
    int m = m0 + r + ((lane >> 4) << 3);   // C layout: lanes16-31 -> M+8
    int b = m >> 12;
    int l = m & (L_SEQ - 1);
#pragma unroll
    for (int t = 0; t < 4; ++t) {
      int e = n0 + t * 16 + lr;
      float v = acc[t][r];
      if (e < DI) xbuf[((b * DI + e)        << 12) + l] = v;
      else        zbuf[((b * DI + (e - DI)) << 12) + l] = v;
    }
  }
}

// ---------------------------------------------------------------------------
// Depthwise causal 4-tap conv + bias + SiLU
// ---------------------------------------------------------------------------
__global__ __launch_bounds__(256) void conv_silu_kernel(
    const float* __restrict__ x, const float* __restrict__ cw,
    const float* __restrict__ cb, float* __restrict__ u)
{
  int idx  = blockIdx.x * 256 + threadIdx.x;
  int l    = idx & (L_SEQ - 1);
  int c    = (idx >> 12) & (DI - 1);
  int base = idx - l;
  float acc = cb[c];
#pragma unroll
  for (int j = 0; j < 4; ++j) {
    int ll = l + j - 3;
    if (ll >= 0) acc += cw[(c << 2) + j] * x[base + ll];
  }
  u[idx] = acc / (1.0f + __expf(-acc));
}

// ---------------------------------------------------------------------------
// FFT helpers (LDS twiddle table: tw[j] = exp(-i*pi*j/4096), j in [0,4096))
// ---------------------------------------------------------------------------
__device__ __forceinline__ float2 cmul(float2 a, float2 b) {
  return make_float2(a.x * b.x - a.y * b.y, a.x * b.y + a.y * b.x);
}

__device__ __forceinline__ void build_twiddles(float2* tw, int tid) {
  for (int j = tid; j < (NFFT >> 1); j += 512) {
    float ang = -(float)M_PI * (float)j * (1.0f / 4096.0f);
    tw[j] = make_float2(__cosf(ang), __sinf(ang));
  }
}

// Forward radix-2 DIF: natural order in, BIT-REVERSED order out.
__device__ __forceinline__ void fft_dif_fwd(float2* sh, const float2* tw, int tid) {
  int sa = 0;                                   // tw index shift: j << sa
  for (int half = NFFT >> 1; half >= 1; half >>= 1, ++sa) {
    for (int p = tid; p < (NFFT >> 1); p += 512) {
      int j  = p & (half - 1);
      int i0 = ((p & ~(half - 1)) << 1) | j;
      int i1 = i0 + half;
      float2 x0 = sh[i0], x1 = sh[i1];
      float2 s  = make_float2(x0.x + x1.x, x0.y + x1.y);
      float2 d  = make_float2(x0.x - x1.x, x0.y - x1.y);
      float2 w  = tw[j << sa];                  // exp(-i*pi*j/half)
      sh[i0] = s;
      sh[i1] = cmul(d, w);
    }
    __syncthreads();
  }
}

// Inverse radix-2 DIT: BIT-REVERSED in, natural out (conjugate twiddles).
__device__ __forceinline__ void fft_dit_inv(float2* sh, const float2* tw, int tid) {
  int sa = 12;
  for (int half = 1; half <= (NFFT >> 1); half <<= 1, --sa) {
    for (int p = tid; p < (NFFT >> 1); p += 512) {
      int j  = p & (half - 1);
      int i0 = ((p & ~(half - 1)) << 1) | j;
      int i1 = i0 + half;
      float2 wt = tw[j << sa];
      float2 w  = make_float2(wt.x, -wt.y);     // conjugate -> exp(+i*pi*j/half)
      float2 t  = cmul(w, sh[i1]);
      float2 x0 = sh[i0];
      sh[i1] = make_float2(x0.x - t.x, x0.y - t.y);
      sh[i0] = make_float2(x0.x + t.x, x0.y + t.y);
    }
    __syncthreads();
  }
}

// ---------------------------------------------------------------------------
// K_f = DIF-FFT(k_ssm zero-padded to 8192), stored bit-reversed. One WG/ch.
// Dynamic LDS: sh[8192] (64KB) + tw[4096] (32KB)
// ---------------------------------------------------------------------------
__global__ __launch_bounds__(512) void fft_k_kernel(
    const float* __restrict__ kssm, float2* __restrict__ kf)
{
  extern __shared__ float2 smem_k[];
  float2* sh = smem_k;
  float2* tw = smem_k + NFFT;
  const int tid = threadIdx.x;
  const int c   = blockIdx.x;
  for (int i = tid; i < NFFT; i += 512) {
    float re = (i < L_SEQ) ? kssm[c * L_SEQ + i] : 0.0f;
    sh[i] = make_float2(re, 0.0f);
  }
  build_twiddles(tw, tid);
  __syncthreads();
  fft_dif_fwd(sh, tw, tid);
  for (int i = tid; i < NFFT; i += 512) kf[c * NFFT + i] = sh[i];
}

// ---------------------------------------------------------------------------
// Per (b,c): TDM-prefetch K_f[c] into LDS, FFT(u), *K_f (bit-reversed),
// IFFT, +D*u, *silu(z).  Dynamic LDS: sh 64KB | kfl 64KB | tw 32KB = 160KB
// (2 workgroups per 320KB WGP).
// ---------------------------------------------------------------------------
__global__ __launch_bounds__(512) void fftconv_kernel(
    const float* __restrict__ u, const float* __restrict__ z,
    const float2* __restrict__ kf, const float* __restrict__ Dskip,
    float* __restrict__ ybuf)
{
  extern __shared__ float2 smem_c[];
  float2* sh  = smem_c;           // 8192 complex : data
  float2* kfl = smem_c + NFFT;    // 8192 complex : channel spectrum
  float2* tw  = kfl + NFFT;       // 4096 complex : twiddles

  const int tid   = threadIdx.x;
  const int bc    = blockIdx.x;          // b*DI + c
  const int c     = bc & (DI - 1);
  const int ubase = bc << 12;

  // --- Kick off async TDM copy of K_f[c] (64KB) global -> LDS (wave 0 only;
  //     TENSORcnt is per-wave).  1-D descriptor: 8192 x 8B elements.
  if (threadIdx.x < 32) {
    unsigned long long ga = (unsigned long long)(uintptr_t)(kf + (size_t)c * NFFT);
    unsigned int lds_off  = (unsigned int)(uintptr_t)kfl;   // addr[31:0] = LDS offset
    u32x4 g0;
    g0[0] = 1u;                                             // count=1, user mode
    g0[1] = lds_off;                                        // lds_addr
    g0[2] = (unsigned int)ga;                               // global_addr[31:0]
    g0[3] = ((unsigned int)(ga >> 32) & 0x01FFFFFFu)        // global_addr[56:32]
            | 0x80000000u;                                  // type=2 (bits 127:126)
    i32x8 g1;
    g1[0] = (int)(3u << 16);          // wg_mask=0, data_size=3 (8B)
    g1[1] = (int)(0x2000u << 16);     // tensor_dim0[15:0]=8192 at bits[31:16]
    g1[2] = (int)(1u << 16);          // tensor_dim0 hi=0, tensor_dim1=1
    g1[3] = (int)(0x2000u << 16);     // tensor_dim1 hi=0, tile_dim0=8192
    g1[4] = 1;                        // tile_dim1=1, tile_dim2=0
    g1[5] = 0x2000;                   // tensor_dim0_stride=8192
    g1[6] = 0;
    g1[7] = 0;
    i32x4 gz4 = {0, 0, 0, 0};
    i32x8 gz8 = {0, 0, 0, 0, 0, 0, 0, 0};
    __builtin_amdgcn_tensor_load_to_lds(g0, g1, gz4, gz4, gz8, 0);
  }

  // --- Load u (zero-padded) while the TDM runs
  for (int i = tid; i < NFFT; i += 512) {
    float re = (i < L_SEQ) ? u[ubase + i] : 0.0f;
    sh[i] = make_float2(re, 0.0f);
  }
  build_twiddles(tw, tid);
  __syncthreads();

  fft_dif_fwd(sh, tw, tid);            // overlapped with TDM transfer

  if (threadIdx.x < 32) __builtin_amdgcn_s_wait_tensorcnt(0);
  __syncthreads();

  for (int i = tid; i < NFFT; i += 512)
    sh[i] = cmul(sh[i], kfl[i]);
  __syncthreads();

  fft_dit_inv(sh, tw, tid);

  const float ds  = Dskip[c];
  const float inv = 1.0f / (float)NFFT;
  for (int l = tid; l < L_SEQ; l += 512) {
    float conv = sh[l].x * inv;
    float uu   = u[ubase + l];
    float zz   = z[ubase + l];
    float yy   = conv + ds * uu;
    float sz   = zz / (1.0f + __expf(-zz));
    ybuf[ubase + l] = yy * sz;
  }
}

// ---------------------------------------------------------------------------
// GEMM 2: out[b,l,d] = sum_h y[b,h,l] * Wout[d,h]
// ---------------------------------------------------------------------------
__global__ __launch_bounds__(256) void gemm_out_kernel(
    const float* __restrict__ ybuf, const float* __restrict__ Wout,
    float* __restrict__ out)
{
  const int lane = threadIdx.x & 31;
  const int wave = threadIdx.x >> 5;
  const int gw   = blockIdx.x * 8 + wave;
  const int ng   = gw & 15;            // 1024/64 = 16 n-groups
  const int mt   = gw >> 4;            // 512 m tiles
  const int m0   = mt << 4;
  const int n0   = ng << 6;
  const int koff = (lane >> 4) << 1;
  const int lr   = lane & 15;

  const int mrow = m0 + lr;
  const int bb   = mrow >> 12;
  const int ll   = mrow & (L_SEQ - 1);
  const float* pa = ybuf + ((bb * DI) << 12) + ll;  // + k*4096
  const float* pb = Wout + (n0 + lr) * DI + koff;
  __builtin_prefetch(pb, 0, 0);

  v8f acc[4] = {v8f{}, v8f{}, v8f{}, v8f{}};
  for (int k0 = 0; k0 < DI; k0 += 4) {
    int ka = k0 + koff;
    v2f a;
    a.x = pa[ka << 12];
    a.y = pa[(ka + 1) << 12];
#pragma unroll
    for (int t = 0; t < 4; ++t) {
      v2f b = *(const v2f*)(pb + t * 16 * DI + k0);
      acc[t] = __builtin_amdgcn_wmma_f32_16x16x4_f32(
          false, a, false, b, (short)0, acc[t], false, false);
    }
  }
#pragma unroll
  for (int r = 0; r < 8; ++r) {
    int m = m0 + r + ((lane >> 4) << 3);
#pragma unroll
    for (int t = 0; t < 4; ++t) {
      int d = n0 + t * 16 + lr;
      out[m * DM + d] = acc[t][r];
    }
  }
}

// ---------------------------------------------------------------------------
// Host launch
// ---------------------------------------------------------------------------
extern "C" void kernel_launch(void* const* d_in, const int* in_sizes, int n_in,
                              void* d_out, int out_size, void* d_ws, size_t ws_size,
                              hipStream_t stream) {
  const float* hs    = (const float*)d_in[0];  // (B, L, DM)
  const float* Win   = (const float*)d_in[1];  // (2*DI, DM)
  const float* convw = (const float*)d_in[2];  // (DI, 4)
  const float* convb = (const float*)d_in[3];  // (DI,)
  const float* kssm  = (const float*)d_in[4];  // (DI, L)
  const float* Dskip = (const float*)d_in[5];  // (DI,)
  const float* Wout  = (const float*)d_in[6];  // (DM, DI)
  float* out = (float*)d_out;                  // (B, L, DM)

  const size_t NCH = (size_t)NBATCH * DI * L_SEQ; // 16,777,216 floats
  float*  xbuf = (float*)d_ws;
  float*  zbuf = xbuf + NCH;
  float*  ubuf = zbuf + NCH;
  float2* kf   = (float2*)(ubuf + NCH);          // (DI, NFFT) complex

  const int LDS_K  = (NFFT + NFFT / 2) * sizeof(float2);           //  96 KB
  const int LDS_CV = (NFFT + NFFT + NFFT / 2) * sizeof(float2);    // 160 KB
  (void)hipFuncSetAttribute(reinterpret_cast<const void*>(fft_k_kernel),
                            hipFuncAttributeMaxDynamicSharedMemorySize, LDS_K);
  (void)hipFuncSetAttribute(reinterpret_cast<const void*>(fftconv_kernel),
                            hipFuncAttributeMaxDynamicSharedMemorySize, LDS_CV);

  // 1) input projection (WMMA fp32), writes x/z channel-major
  gemm_xz_kernel<<<4096, 256, 0, stream>>>(hs, Win, xbuf, zbuf);
  // 2) depthwise conv + SiLU
  conv_silu_kernel<<<(NBATCH * DI * L_SEQ) / 256, 256, 0, stream>>>(
      xbuf, convw, convb, ubuf);
  // 3) kernel spectrum (bit-reversed DIF)
  fft_k_kernel<<<DI, 512, LDS_K, stream>>>(kssm, kf);
  // 4) FFT conv + skip + gating (TDM-prefetched spectrum); y overwrites xbuf
  fftconv_kernel<<<NBATCH * DI, 512, LDS_CV, stream>>>(ubuf, zbuf, kf, Dskip, xbuf);
  // 5) output projection (WMMA fp32)
  gemm_out_kernel<<<1024, 256, 0, stream>>>(xbuf, Wout, out);
}